// GIN0WithJK_61229053772418
// MI455X (gfx1250) — compile-verified
//
#include <hip/hip_runtime.h>

#define N_NODESC 50000
#define N_EDGESC 600000
#define HIDC 128
#define N_OUTC 8
#define N_LAYERSC 4
#define BN_EPSF 1e-5f

typedef __attribute__((ext_vector_type(2))) float v2f;
typedef __attribute__((ext_vector_type(8))) float v8f;

#define KCHUNK 128
#define LDS_PAD 4
#define ASTRIDE (128 + LDS_PAD)

// ---------------------------------------------------------------------------
// Zero-fill (float4)
// ---------------------------------------------------------------------------
__global__ __launch_bounds__(256) void zero_kernel(float4* __restrict__ p, int n4) {
    int i = blockIdx.x * 256 + threadIdx.x;
    if (i < n4) p[i] = make_float4(0.f, 0.f, 0.f, 0.f);
}

// ---------------------------------------------------------------------------
// Edge scatter: agg[dst] += h[src]   (32 lanes per edge, 4 floats per lane)
// ---------------------------------------------------------------------------
__global__ __launch_bounds__(256) void scatter_kernel(
    const float* __restrict__ h, int ldh,
    const int* __restrict__ src, const int* __restrict__ dst,
    float* __restrict__ agg) {
    int gid = blockIdx.x * 256 + threadIdx.x;
    int e = gid >> 5;
    if (e >= N_EDGESC) return;
    int c = (gid & 31) << 2;
    int s = src[e];
    int d = dst[e];
    const float4 v = *(const float4*)(h + (size_t)s * ldh + c);
    float* a = agg + (size_t)d * HIDC + c;
    unsafeAtomicAdd(a + 0, v.x);
    unsafeAtomicAdd(a + 1, v.y);
    unsafeAtomicAdd(a + 2, v.z);
    unsafeAtomicAdd(a + 3, v.w);
}

// ---------------------------------------------------------------------------
// WMMA GEMM: C[:, col0: col0+NT*16] = act(A[nrows,KDIM] @ W[KDIM,ncols] + b)
//   block = 256 threads = 8 waves; block tile = 128 rows x NT*16 cols
//   per wave: 16 rows, NT 16x16 f32 accum tiles, v_wmma_f32_16x16x4_f32
// ---------------------------------------------------------------------------
template <int KDIM, int NT, bool FUSE_ADD, bool RELU>
__global__ __launch_bounds__(256) void gemm_kernel(
    const float* __restrict__ A, int lda,
    const float* __restrict__ Aadd,             // [nrows,128], used iff FUSE_ADD
    const float* __restrict__ W, int ldw, int ncols,
    const float* __restrict__ bias,
    float* __restrict__ C, int ldc, int col0,
    int nrows) {
    constexpr int BST = NT * 16 + LDS_PAD;
    __shared__ float ldsA[KCHUNK * ASTRIDE];
    __shared__ float ldsB[KCHUNK * BST];

    const int tid = threadIdx.x;
    const int wave = tid >> 5;
    const int lane = tid & 31;
    const int row0 = blockIdx.x * 128;

    v8f acc[NT];
#pragma unroll
    for (int t = 0; t < NT; ++t) acc[t] = (v8f){0.f, 0.f, 0.f, 0.f, 0.f, 0.f, 0.f, 0.f};

    for (int kc = 0; kc < KDIM; kc += KCHUNK) {
        if (kc) __syncthreads();
        // Stage A chunk: 128 rows x 128 K (float4, clamped rows for tail block)
        for (int idx = tid; idx < 128 * 32; idx += 256) {
            int r = idx >> 5;
            int c4 = (idx & 31) << 2;
            int row = row0 + r;
            if (row >= nrows) row = nrows - 1;
            float4 v = *(const float4*)(A + (size_t)row * lda + kc + c4);
            if (FUSE_ADD) {
                const float4 g = *(const float4*)(Aadd + (size_t)row * HIDC + c4);
                v.x += g.x; v.y += g.y; v.z += g.z; v.w += g.w;
            }
            float* d = &ldsA[r * ASTRIDE + c4];
            d[0] = v.x; d[1] = v.y; d[2] = v.z; d[3] = v.w;
        }
        // Stage B chunk: 128 K x NT*16 N (zero-pad beyond ncols)
        for (int idx = tid; idx < KCHUNK * NT * 16; idx += 256) {
            int k = idx / (NT * 16);
            int n = idx % (NT * 16);
            ldsB[k * BST + n] = (n < ncols) ? W[(size_t)(kc + k) * ldw + n] : 0.f;
        }
        __syncthreads();

        // A frag: lanes 0-15 -> K = kk,kk+1 ; lanes 16-31 -> K = kk+2,kk+3
        const int mrow = wave * 16 + (lane & 15);
        const int khalf = (lane >> 4) << 1;
        for (int kk = 0; kk < KCHUNK; kk += 4) {
            v2f a;
            a.x = ldsA[mrow * ASTRIDE + kk + khalf];
            a.y = ldsA[mrow * ASTRIDE + kk + khalf + 1];
#pragma unroll
            for (int t = 0; t < NT; ++t) {
                const int n = t * 16 + (lane & 15);
                v2f b;
                b.x = ldsB[(kk + khalf) * BST + n];
                b.y = ldsB[(kk + khalf + 1) * BST + n];
                acc[t] = __builtin_amdgcn_wmma_f32_16x16x4_f32(
                    false, a, false, b, (short)0, acc[t], false, false);
            }
        }
    }

    // Epilogue: bias + activation + guarded store
    // D layout: VGPR r, lanes 0-15 -> M=r, lanes 16-31 -> M=8+r; N = lane&15
    const int rbase = row0 + wave * 16 + ((lane >> 4) << 3);
#pragma unroll
    for (int t = 0; t < NT; ++t) {
        const int col = t * 16 + (lane & 15);
        const float bv = (col < ncols) ? bias[col] : 0.f;
#pragma unroll
        for (int r = 0; r < 8; ++r) {
            const int row = rbase + r;
            float v = acc[t][r] + bv;
            if (RELU) v = fmaxf(v, 0.f);
            if (row < nrows && col < ncols)
                C[(size_t)row * ldc + col0 + col] = v;
        }
    }
}

// ---------------------------------------------------------------------------
// BatchNorm: deterministic two-pass stats + apply into strided JK buffer
// ---------------------------------------------------------------------------
#define BN_BLOCKS 250
#define BN_ROWS (N_NODESC / BN_BLOCKS)  // 200

__global__ __launch_bounds__(128) void bn_partial_kernel(
    const float* __restrict__ h, float* __restrict__ partial) {
    const int c = threadIdx.x;
    const int r0 = blockIdx.x * BN_ROWS;
    float s = 0.f, ss = 0.f;
    for (int r = r0; r < r0 + BN_ROWS; ++r) {
        const float v = h[(size_t)r * HIDC + c];
        s += v;
        ss += v * v;
    }
    partial[blockIdx.x * 256 + c] = s;
    partial[blockIdx.x * 256 + 128 + c] = ss;
}

__global__ __launch_bounds__(128) void bn_finalize_kernel(
    const float* __restrict__ partial,
    const float* __restrict__ gamma, const float* __restrict__ beta,
    float* __restrict__ scsh) {
    const int c = threadIdx.x;
    float s = 0.f, ss = 0.f;
    for (int b = 0; b < BN_BLOCKS; ++b) {
        s += partial[b * 256 + c];
        ss += partial[b * 256 + 128 + c];
    }
    const float inv_n = 1.0f / (float)N_NODESC;
    const float mean = s * inv_n;
    const float var = ss * inv_n - mean * mean;
    const float sc = gamma[c] * rsqrtf(var + BN_EPSF);
    scsh[c] = sc;
    scsh[128 + c] = beta[c] - mean * sc;
}

__global__ __launch_bounds__(256) void bn_apply_kernel(
    const float* __restrict__ h, const float* __restrict__ scsh,
    float* __restrict__ jk, int col0) {
    const int gid = blockIdx.x * 256 + threadIdx.x;  // over N*128
    const int row = gid >> 7;
    const int c = gid & 127;
    const float v = h[gid];
    jk[(size_t)row * (N_LAYERSC * HIDC) + col0 + c] = v * scsh[c] + scsh[128 + c];
}

// ---------------------------------------------------------------------------
// Host-side orchestration
// ---------------------------------------------------------------------------
extern "C" void kernel_launch(void* const* d_in, const int* in_sizes, int n_in,
                              void* d_out, int out_size, void* d_ws, size_t ws_size,
                              hipStream_t stream) {
    const float* x     = (const float*)d_in[0];
    const int*   edge  = (const int*)d_in[1];
    const float* W1    = (const float*)d_in[2];
    const float* b1    = (const float*)d_in[3];
    const float* W2    = (const float*)d_in[4];
    const float* b2    = (const float*)d_in[5];
    const float* gamma = (const float*)d_in[6];
    const float* beta  = (const float*)d_in[7];
    const float* l1W   = (const float*)d_in[8];
    const float* l1b   = (const float*)d_in[9];
    const float* l2W   = (const float*)d_in[10];
    const float* l2b   = (const float*)d_in[11];
    const int* src = edge;
    const int* dst = edge + N_EDGESC;
    float* out = (float*)d_out;

    float* ws   = (float*)d_ws;
    float* jk   = ws;                               // [N, 512]
    float* agg  = jk  + (size_t)N_NODESC * 512;     // [N, 128]
    float* t1   = agg + (size_t)N_NODESC * HIDC;    // [N, 128]
    float* t2   = t1  + (size_t)N_NODESC * HIDC;    // [N, 128]
    float* part = t2  + (size_t)N_NODESC * HIDC;    // [250, 256]
    float* scsh = part + BN_BLOCKS * 256;           // [256]

    const int gemm_blocks = (N_NODESC + 127) / 128;          // 391
    const int zero_blocks = (N_NODESC * HIDC / 4) / 256;     // 6250
    const int scat_blocks = (N_EDGESC * 32) / 256;           // 75000
    const int appl_blocks = (N_NODESC * HIDC) / 256;         // 25000

    for (int l = 0; l < N_LAYERSC; ++l) {
        const float* hprev = (l == 0) ? x : (jk + (size_t)(l - 1) * HIDC);
        const int ldprev = (l == 0) ? HIDC : (N_LAYERSC * HIDC);

        zero_kernel<<<zero_blocks, 256, 0, stream>>>((float4*)agg, N_NODESC * HIDC / 4);
        scatter_kernel<<<scat_blocks, 256, 0, stream>>>(hprev, ldprev, src, dst, agg);

        // t1 = relu((hprev + agg) @ W1[l] + b1[l])
        gemm_kernel<128, 8, true, true><<<gemm_blocks, 256, 0, stream>>>(
            hprev, ldprev, agg,
            W1 + (size_t)l * HIDC * HIDC, HIDC, HIDC, b1 + (size_t)l * HIDC,
            t1, HIDC, 0, N_NODESC);

        // t2 = relu(t1 @ W2[l] + b2[l])
        gemm_kernel<128, 8, false, true><<<gemm_blocks, 256, 0, stream>>>(
            t1, HIDC, nullptr,
            W2 + (size_t)l * HIDC * HIDC, HIDC, HIDC, b2 + (size_t)l * HIDC,
            t2, HIDC, 0, N_NODESC);

        // BatchNorm -> jk[:, l*128:(l+1)*128]
        bn_partial_kernel<<<BN_BLOCKS, 128, 0, stream>>>(t2, part);
        bn_finalize_kernel<<<1, 128, 0, stream>>>(
            part, gamma + (size_t)l * HIDC, beta + (size_t)l * HIDC, scsh);
        bn_apply_kernel<<<appl_blocks, 256, 0, stream>>>(t2, scsh, jk, l * HIDC);
    }

    // t1 = relu(jk @ lin1_W + lin1_b)   ([N,512] x [512,128])
    gemm_kernel<512, 8, false, true><<<gemm_blocks, 256, 0, stream>>>(
        jk, N_LAYERSC * HIDC, nullptr, l1W, HIDC, HIDC, l1b, t1, HIDC, 0, N_NODESC);

    // out = t1 @ lin2_W + lin2_b        ([N,128] x [128,8])
    gemm_kernel<128, 1, false, false><<<gemm_blocks, 256, 0, stream>>>(
        t1, HIDC, nullptr, l2W, N_OUTC, N_OUTC, l2b, out, N_OUTC, 0, N_NODESC);
}